// RiemannianBatchNormSPD_84095459656162
// MI455X (gfx1250) — compile-verified
//
#include <hip/hip_runtime.h>
#include <hip/hip_bf16.h>
#include <math.h>

// ---------------------------------------------------------------------------
// RiemannianBatchNormSPD for 32x32 SPD matrices, matmul-only log/exp maps.
// One wave32 = one matrix, held as 2x2 grid of 16x16 WMMA C/D-layout v8f
// fragments. All matmuls use V_WMMA_F32_16X16X4_F32. Operands are staged
// row-major in per-wave LDS buffers; since every operand in this algorithm
// is symmetric (polynomials in one SPD matrix), BOTH A- and B-fragments are
// contiguous ds_load_b64 reads from the same row-major image.
// ---------------------------------------------------------------------------

typedef float v2f __attribute__((ext_vector_type(2)));
typedef float v8f __attribute__((ext_vector_type(8)));

#define WAVES  4      // waves (matrices) per block
#define NSQRT  5      // inverse-scaling-and-squaring depth for logm
#define NSITER 14     // Newton-Schulz iterations per square root
#define LOGORD 8      // Gregory series order for log(I+E)

struct Mat32 { v8f b[2][2]; };

// C/D layout: VGPR r of block (bi,bj) holds row M = bi*16 + r + 8*hi, col N = bj*16 + lo
__device__ __forceinline__ int midx(int bi, int bj, int r, int lo, int hi) {
  return (bi * 16 + r + hi * 8) * 32 + bj * 16 + lo;
}

__device__ __forceinline__ void mat_store(float* dst, const Mat32& M, int lo, int hi) {
#pragma unroll
  for (int bi = 0; bi < 2; ++bi)
#pragma unroll
    for (int bj = 0; bj < 2; ++bj)
#pragma unroll
      for (int r = 0; r < 8; ++r)
        dst[midx(bi, bj, r, lo, hi)] = M.b[bi][bj][r];
}

// C = A * B for SYMMETRIC operands staged row-major (32x32 f32) in LDS.
// A-frag (16x4, k-step s): lane holds A[bi*16+lo][k .. k+1]        (contiguous b64)
// B-frag (4x16, k-step s): lane needs B[k+r][bj*16+lo] = B[bj*16+lo][k+r] (contiguous b64)
__device__ __forceinline__ Mat32 lds_mm_sym(const float* lA, const float* lB, int lo, int hi) {
  v2f af[2][8];
  v2f bf[2][8];
#pragma unroll
  for (int bi = 0; bi < 2; ++bi)
#pragma unroll
    for (int kb = 0; kb < 2; ++kb)
#pragma unroll
      for (int s = 0; s < 4; ++s)
        af[bi][kb * 4 + s] =
            *(const v2f*)(lA + (bi * 16 + lo) * 32 + kb * 16 + 4 * s + 2 * hi);
#pragma unroll
  for (int bj = 0; bj < 2; ++bj)
#pragma unroll
    for (int kb = 0; kb < 2; ++kb)
#pragma unroll
      for (int s = 0; s < 4; ++s)
        bf[bj][kb * 4 + s] =
            *(const v2f*)(lB + (bj * 16 + lo) * 32 + kb * 16 + 4 * s + 2 * hi);
  Mat32 C;
#pragma unroll
  for (int bi = 0; bi < 2; ++bi)
#pragma unroll
    for (int bj = 0; bj < 2; ++bj) {
      v8f acc = {};
#pragma unroll
      for (int t = 0; t < 8; ++t)
        acc = __builtin_amdgcn_wmma_f32_16x16x4_f32(
            false, af[bi][t], false, bf[bj][t], (short)0, acc, false, false);
      C.b[bi][bj] = acc;
    }
  return C;
}

// D = a*S + dg*I  (elementwise in C/D layout; diagonal iff bi==bj && r+8*hi==lo)
__device__ __forceinline__ Mat32 axpbI(const Mat32& S, float a, float dg, int lo, int hi) {
  Mat32 D;
#pragma unroll
  for (int bi = 0; bi < 2; ++bi)
#pragma unroll
    for (int bj = 0; bj < 2; ++bj)
#pragma unroll
      for (int r = 0; r < 8; ++r) {
        float id = (bi == bj && (r + hi * 8) == lo) ? dg : 0.0f;
        D.b[bi][bj][r] = a * S.b[bi][bj][r] + id;
      }
  return D;
}

__device__ __forceinline__ float wave_sum32(float v) {
#pragma unroll
  for (int off = 16; off > 0; off >>= 1) v += __shfl_xor(v, off, 32);
  return v;
}

__device__ __forceinline__ float mat_trace(const Mat32& M, int lo, int hi) {
  float t = 0.0f;
#pragma unroll
  for (int bi = 0; bi < 2; ++bi)
#pragma unroll
    for (int r = 0; r < 8; ++r)
      if ((r + hi * 8) == lo) t += M.b[bi][bi][r];
  return wave_sum32(t);
}

__device__ __forceinline__ float mat_fro2(const Mat32& M) {
  float s = 0.0f;
#pragma unroll
  for (int bi = 0; bi < 2; ++bi)
#pragma unroll
    for (int bj = 0; bj < 2; ++bj)
#pragma unroll
      for (int r = 0; r < 8; ++r) {
        float x = M.b[bi][bj][r];
        s += x * x;
      }
  return wave_sum32(s);
}

// ---------------------------------------------------------------------------
__global__ void spd_zero_kernel(float* __restrict__ p, int n) {
  int i = blockIdx.x * blockDim.x + threadIdx.x;
  if (i < n) p[i] = 0.0f;
}

// Phase 1: L = logm(X) per matrix; write L to Lout; block-reduce sum(L) -> global accum
__global__ __launch_bounds__(WAVES * 32, 1)
void spd_logmap_kernel(const float* __restrict__ X, float* __restrict__ Lout,
                       float* __restrict__ accum) {
  __shared__ float lds[WAVES * 3072];   // 3 per-wave 32x32 staging buffers
  __shared__ float accLDS[1024];
  const int tid  = threadIdx.x;
  const int lane = tid & 31;
  const int w    = tid >> 5;
  const int lo   = lane & 15;
  const int hi   = lane >> 4;
  float* b0 = lds + w * 3072;
  float* b1 = b0 + 1024;
  float* b2 = b0 + 2048;

  for (int i = tid; i < 1024; i += WAVES * 32) accLDS[i] = 0.0f;
  __syncthreads();

  const size_t n  = (size_t)blockIdx.x * WAVES + w;
  const float* Xn = X + n * 1024;

  Mat32 A;
#pragma unroll
  for (int bi = 0; bi < 2; ++bi)
#pragma unroll
    for (int bj = 0; bj < 2; ++bj)
#pragma unroll
      for (int r = 0; r < 8; ++r)
        A.b[bi][bj][r] = Xn[midx(bi, bj, r, lo, hi)];

  // scale so spectrum in (0,1]: trace bounds spectral norm for SPD
  const float c    = fmaxf(mat_trace(A, lo, hi), 1e-30f);
  const float invc = 1.0f / c;
  Mat32 Acur = axpbI(A, invc, 0.0f, lo, hi);

  // repeated matrix square roots via coupled Newton-Schulz (3 matmuls/iter)
#pragma unroll 1
  for (int rq = 0; rq < NSQRT; ++rq) {
    Mat32 Y = Acur;
    Mat32 Z = axpbI(Y, 0.0f, 1.0f, lo, hi);  // identity
#pragma unroll 1
    for (int it = 0; it < NSITER; ++it) {
      mat_store(b0, Y, lo, hi);
      mat_store(b1, Z, lo, hi);
      Mat32 T = lds_mm_sym(b1, b0, lo, hi);        // Z*Y
      Mat32 W = axpbI(T, -0.5f, 1.5f, lo, hi);     // (3I - ZY)/2
      mat_store(b2, W, lo, hi);
      Y = lds_mm_sym(b0, b2, lo, hi);              // Y*W
      Z = lds_mm_sym(b2, b1, lo, hi);              // W*Z
    }
    Acur = Y;                                      // Acur^(1/2)
  }

  // log(I+E) = E * q(E), q = sum_{j=0..7} (-1)^j E^j/(j+1)   (Horner)
  Mat32 E = axpbI(Acur, 1.0f, -1.0f, lo, hi);
  mat_store(b0, E, lo, hi);                        // E is loop-invariant
  Mat32 S = axpbI(E, 0.0f, -1.0f / (float)LOGORD, lo, hi);
#pragma unroll 1
  for (int j = LOGORD - 2; j >= 0; --j) {
    mat_store(b1, S, lo, hi);
    Mat32 P = lds_mm_sym(b0, b1, lo, hi);          // E*S
    float dj = ((j & 1) ? -1.0f : 1.0f) / (float)(j + 1);
    S = axpbI(P, 1.0f, dj, lo, hi);
  }
  mat_store(b1, S, lo, hi);
  Mat32 LG = lds_mm_sym(b0, b1, lo, hi);           // log(Acur)

  const float sc = (float)(1 << NSQRT);            // undo square roots
  Mat32 L = axpbI(LG, sc, logf(c), lo, hi);        // + log(c)*I undoes scaling

  float* On = Lout + n * 1024;
#pragma unroll
  for (int bi = 0; bi < 2; ++bi)
#pragma unroll
    for (int bj = 0; bj < 2; ++bj)
#pragma unroll
      for (int r = 0; r < 8; ++r) {
        int id  = midx(bi, bj, r, lo, hi);
        float v = L.b[bi][bj][r];
        On[id] = v;
        atomicAdd(&accLDS[id], v);                 // ds_add_f32, cross-wave
      }
  __syncthreads();
  for (int i = tid; i < 1024; i += WAVES * 32)
    __hip_atomic_fetch_add(&accum[i], accLDS[i], __ATOMIC_RELAXED,
                           __HIP_MEMORY_SCOPE_AGENT);
}

// Phase 2: Lc = gamma*(L - mean) + beta; Y = expm(Lc) (scaling & squaring), in-place
__global__ __launch_bounds__(WAVES * 32, 1)
void spd_expmap_kernel(float* __restrict__ InOut, const float* __restrict__ accum,
                       const float* __restrict__ gamma, const float* __restrict__ beta,
                       int BK) {
  __shared__ float lds[WAVES * 2048];   // 2 per-wave staging buffers (R, S)
  const int tid  = threadIdx.x;
  const int lane = tid & 31;
  const int w    = tid >> 5;
  const int lo   = lane & 15;
  const int hi   = lane >> 4;
  float* b0 = lds + w * 2048;
  float* b1 = b0 + 1024;

  const size_t n = (size_t)blockIdx.x * WAVES + w;
  float* Pn = InOut + n * 1024;
  const float g     = gamma[0];
  const float be    = beta[0];
  const float invBK = 1.0f / (float)BK;

  Mat32 R;
#pragma unroll
  for (int bi = 0; bi < 2; ++bi)
#pragma unroll
    for (int bj = 0; bj < 2; ++bj)
#pragma unroll
      for (int r = 0; r < 8; ++r) {
        int id = midx(bi, bj, r, lo, hi);
        R.b[bi][bj][r] = g * (Pn[id] - accum[id] * invBK) + be;
      }

  // pick s so ||R/2^s||_F <= 0.25 (wave-uniform)
  float fro = sqrtf(mat_fro2(R));
  int sh = 0;
  float t = fro;
  while (t > 0.25f && sh < 30) { t *= 0.5f; ++sh; }
  R = axpbI(R, ldexpf(1.0f, -sh), 0.0f, lo, hi);
  mat_store(b0, R, lo, hi);                        // R is loop-invariant

  // exp(R) ~= sum_{k=0..8} R^k/k!  via Horner (8 matmuls)
  const float invfact[8] = {1.0f, 1.0f, 0.5f, 1.0f / 6.0f, 1.0f / 24.0f,
                            1.0f / 120.0f, 1.0f / 720.0f, 1.0f / 5040.0f};
  Mat32 S = axpbI(R, 0.0f, 1.0f / 40320.0f, lo, hi);
#pragma unroll 1
  for (int k = 7; k >= 0; --k) {
    mat_store(b1, S, lo, hi);
    Mat32 P = lds_mm_sym(b0, b1, lo, hi);          // R*S
    S = axpbI(P, 1.0f, invfact[k], lo, hi);
  }
  // repeated squaring (wave-uniform trip count)
#pragma unroll 1
  for (int q = 0; q < sh; ++q) {
    mat_store(b1, S, lo, hi);
    S = lds_mm_sym(b1, b1, lo, hi);
  }

#pragma unroll
  for (int bi = 0; bi < 2; ++bi)
#pragma unroll
    for (int bj = 0; bj < 2; ++bj)
#pragma unroll
      for (int r = 0; r < 8; ++r)
        Pn[midx(bi, bj, r, lo, hi)] = S.b[bi][bj][r];
}

// ---------------------------------------------------------------------------
extern "C" void kernel_launch(void* const* d_in, const int* in_sizes, int n_in,
                              void* d_out, int out_size, void* d_ws, size_t ws_size,
                              hipStream_t stream) {
  const float* X     = (const float*)d_in[0];
  const float* gamma = (const float*)d_in[1];
  const float* beta  = (const float*)d_in[2];
  float* out   = (float*)d_out;
  float* accum = (float*)d_ws;            // 1024 floats: sum of L over batch

  const int BK     = in_sizes[0] / 1024;  // number of 32x32 matrices (B*K)
  const int blocks = (BK + WAVES - 1) / WAVES;

  spd_zero_kernel<<<4, 256, 0, stream>>>(accum, 1024);
  spd_logmap_kernel<<<blocks, WAVES * 32, 0, stream>>>(X, out, accum);
  spd_expmap_kernel<<<blocks, WAVES * 32, 0, stream>>>(out, accum, gamma, beta, BK);
}